// HierarchicalProofEncoder_61624190763589
// MI455X (gfx1250) — compile-verified
//
#include <hip/hip_runtime.h>
#include <hip/hip_bf16.h>

typedef float v2f __attribute__((ext_vector_type(2)));
typedef float v8f __attribute__((ext_vector_type(8)));

// ---------------- problem constants (match reference) ----------------
static constexpr int NB   = 64;     // NUM_GRAPHS
static constexpr int NMAX = 512;    // MAX_NODES
static constexpr int DIN  = 256;    // IN_DIM
static constexpr int HIDD = 256;    // HID
static constexpr int C1D  = 64;     // C1
static constexpr int C2D  = 16;     // C2
static constexpr int ODIM = 256;    // OUT_DIM

// =====================================================================
//  WMMA f32 GEMM:  C[b] = op(A[b]) * op(B[b]) (+ bias per column)
//  One wave computes one 16x16 tile, K stepped by 4 via
//  V_WMMA_F32_16X16X4_F32.  M,N multiples of 16; K multiple of 4.
// =====================================================================
template <bool TA, bool TB, bool HB>
__global__ void wmma_gemm_kernel(const float* __restrict__ A,
                                 const float* __restrict__ Bm,
                                 const float* __restrict__ bias,
                                 float* __restrict__ C,
                                 int M, int N, int K,
                                 int lda, int ldb, int ldc,
                                 long sA, long sB, long sC) {
  const int lane = threadIdx.x & 31;
  const int wave = threadIdx.x >> 5;
  const int n0 = (blockIdx.x * (blockDim.x >> 5) + wave) * 16;
  const int m0 = blockIdx.y * 16;
  if (n0 >= N || m0 >= M) return;            // wave-uniform exit (EXEC stays full)
  const long zb = blockIdx.z;
  A  += zb * sA;  Bm += zb * sB;  C += zb * sC;

  const int l15  = lane & 15;
  const int koff = (lane >> 4) << 1;         // 0 or 2
  v8f acc = {};
  for (int k = 0; k < K; k += 4) {
    const int ka = k + koff;
    v2f av, bv;
    if (TA) {
      av.x = A[(long)ka * lda + (m0 + l15)];
      av.y = A[(long)(ka + 1) * lda + (m0 + l15)];
    } else {
      const float* p = A + (long)(m0 + l15) * lda + ka;
      av.x = p[0]; av.y = p[1];
    }
    if (TB) {
      const float* p = Bm + (long)(n0 + l15) * ldb + ka;
      bv.x = p[0]; bv.y = p[1];
    } else {
      bv.x = Bm[(long)ka * ldb + (n0 + l15)];
      bv.y = Bm[(long)(ka + 1) * ldb + (n0 + l15)];
    }
    acc = __builtin_amdgcn_wmma_f32_16x16x4_f32(false, av, false, bv,
                                                (short)0, acc, false, false);
  }
  const float badd = HB ? bias[n0 + l15] : 0.0f;
  const int mrow = m0 + ((lane >> 4) << 3);  // m0 or m0+8
#pragma unroll
  for (int v = 0; v < 8; ++v)
    C[(long)(mrow + v) * ldc + (n0 + l15)] = acc[v] + badd;
}

static void run_gemm(hipStream_t st, bool ta, bool tb,
                     const float* A, const float* B, const float* bias, float* C,
                     int M, int N, int K, int lda, int ldb, int ldc,
                     long sA, long sB, long sC, int batch) {
  dim3 blk(128);
  dim3 grd((unsigned)((N / 16 + 3) / 4), (unsigned)(M / 16), (unsigned)batch);
  const bool hb = (bias != nullptr);
#define GEMM_CASE(TA_, TB_, HB_)                                              \
  wmma_gemm_kernel<TA_, TB_, HB_><<<grd, blk, 0, st>>>(                       \
      A, B, bias, C, M, N, K, lda, ldb, ldc, sA, sB, sC)
  if      (!ta && !tb && !hb) GEMM_CASE(false, false, false);
  else if (!ta && !tb &&  hb) GEMM_CASE(false, false, true);
  else if (!ta &&  tb && !hb) GEMM_CASE(false, true,  false);
  else if (!ta &&  tb &&  hb) GEMM_CASE(false, true,  true);
  else if ( ta && !tb && !hb) GEMM_CASE(true,  false, false);
  else if ( ta && !tb &&  hb) GEMM_CASE(true,  false, true);
  else if ( ta &&  tb && !hb) GEMM_CASE(true,  true,  false);
  else                        GEMM_CASE(true,  true,  true);
#undef GEMM_CASE
}

// =====================================================================
//  Small utility kernels
// =====================================================================
__global__ void fill_f32_kernel(float* p, float v, long n) {
  long i = (long)blockIdx.x * blockDim.x + threadIdx.x;
  if (i < n) p[i] = v;
}
__global__ void fill_i32_kernel(int* p, int v, long n) {
  long i = (long)blockIdx.x * blockDim.x + threadIdx.x;
  if (i < n) p[i] = v;
}
__global__ void count_batch_kernel(const int* __restrict__ batch, int* counts, int n) {
  int i = blockIdx.x * blockDim.x + threadIdx.x;
  if (i < n) atomicAdd(&counts[batch[i]], 1);
}
__global__ void scan64_kernel(const int* __restrict__ counts, int* starts, int nb) {
  if (blockIdx.x == 0 && threadIdx.x == 0) {
    int a = 0;
    for (int g = 0; g < nb; ++g) { starts[g] = a; a += counts[g]; }
  }
}
__global__ void local_kernel(const int* __restrict__ batch, const int* __restrict__ starts,
                             int* local, int n) {
  int i = blockIdx.x * blockDim.x + threadIdx.x;
  if (i < n) local[i] = i - starts[batch[i]];
}
__global__ void count_deg_kernel(const int* __restrict__ dst, int* deg, int ne) {
  int e = blockIdx.x * blockDim.x + threadIdx.x;
  if (e < ne) atomicAdd(&deg[dst[e]], 1);
}
// single-block scan: offs[0]=0; offs[i+1] = sum deg[0..i]
__global__ void scan_offsets_kernel(const int* __restrict__ deg, int* __restrict__ offs, int n) {
  __shared__ int buf[1024];
  __shared__ int carry;
  const int tid = threadIdx.x;
  if (tid == 0) { carry = 0; offs[0] = 0; }
  __syncthreads();
  for (int base = 0; base < n; base += 1024) {
    int i = base + tid;
    buf[tid] = (i < n) ? deg[i] : 0;
    __syncthreads();
    for (int off = 1; off < 1024; off <<= 1) {
      int t = (tid >= off) ? buf[tid - off] : 0;
      __syncthreads();
      buf[tid] += t;
      __syncthreads();
    }
    if (i < n) offs[i + 1] = carry + buf[tid];
    __syncthreads();
    if (tid == 0) carry += buf[1023];
    __syncthreads();
  }
}
__global__ void csr_fill_kernel(const int* __restrict__ src, const int* __restrict__ dst,
                                const int* __restrict__ offs, int* cursor, int* csr_src, int ne) {
  int e = blockIdx.x * blockDim.x + threadIdx.x;
  if (e >= ne) return;
  int d = dst[e];
  int pos = offs[d] + atomicAdd(&cursor[d], 1);
  csr_src[pos] = src[e];
}
__global__ void scatter_rows_kernel(const float* __restrict__ srcv,
                                    const int* __restrict__ batchv,
                                    const int* __restrict__ localv,
                                    float* __restrict__ dstv, int n, int D, int nmax) {
  long t = (long)blockIdx.x * blockDim.x + threadIdx.x;
  if (t >= (long)n * D) return;
  int i = (int)(t / D), d = (int)(t % D);
  dstv[((long)batchv[i] * nmax + localv[i]) * D + d] = srcv[t];
}
// softmax over C==64 per node, scattered into dense [NB, NMAX, 64]
__global__ void softmax_scatter_s_kernel(const float* __restrict__ s_in,
                                         const int* __restrict__ batchv,
                                         const int* __restrict__ localv,
                                         float* __restrict__ s_d, int n, int nmax) {
  const int lane = threadIdx.x & 31;
  const int node = blockIdx.x * (blockDim.x >> 5) + (threadIdx.x >> 5);
  if (node >= n) return;
  float v0 = s_in[(long)node * 64 + lane];
  float v1 = s_in[(long)node * 64 + lane + 32];
  float m = fmaxf(v0, v1);
#pragma unroll
  for (int off = 16; off > 0; off >>= 1) m = fmaxf(m, __shfl_xor(m, off, 32));
  float e0 = __expf(v0 - m), e1 = __expf(v1 - m);
  float s = e0 + e1;
#pragma unroll
  for (int off = 16; off > 0; off >>= 1) s += __shfl_xor(s, off, 32);
  const float inv = 1.0f / s;
  long orow = ((long)batchv[node] * nmax + localv[node]) * 64;
  s_d[orow + lane]      = e0 * inv;
  s_d[orow + lane + 32] = e1 * inv;
}
// plain row softmax (small C), one thread per row
__global__ void softmax_row_kernel(const float* __restrict__ in, float* __restrict__ out,
                                   int rows, int C) {
  int r = blockIdx.x * blockDim.x + threadIdx.x;
  if (r >= rows) return;
  const float* p = in + (long)r * C;
  float* q = out + (long)r * C;
  float m = -1e30f;
  for (int c = 0; c < C; ++c) m = fmaxf(m, p[c]);
  float s = 0.0f;
  for (int c = 0; c < C; ++c) { float e = __expf(p[c] - m); q[c] = e; s += e; }
  float inv = 1.0f / s;
  for (int c = 0; c < C; ++c) q[c] *= inv;
}
// per-graph mean of x rows (contiguous segment via starts/counts)
__global__ void pool_mean_x_kernel(const float* __restrict__ x, const int* __restrict__ starts,
                                   const int* __restrict__ counts, float* __restrict__ emb,
                                   int D, int embStride, int off) {
  int g = blockIdx.x, d = threadIdx.x;
  int s0 = starts[g], c = counts[g];
  float acc = 0.0f;
  for (int i = 0; i < c; ++i) acc += x[(long)(s0 + i) * D + d];
  emb[(long)g * embStride + off + d] = acc / (float)c;
}
__global__ void mean_rows_kernel(const float* __restrict__ xin, float* __restrict__ emb,
                                 int rows, int D, int embStride, int off) {
  int g = blockIdx.x, d = threadIdx.x;
  float acc = 0.0f;
  for (int r = 0; r < rows; ++r) acc += xin[((long)g * rows + r) * D + d];
  emb[(long)g * embStride + off + d] = acc / (float)rows;
}

// =====================================================================
//  Sparse GATv2: one wave per destination node, CSR incoming edges
//  plus implicit self loop; two-pass softmax (max, then exp/accumulate).
//  Next-edge source rows are prefetched (global_prefetch_b8) to hide
//  the dependent csr_src -> gather latency chain.
// =====================================================================
template <int DOUT>
__global__ void gat_sparse_kernel(const float* __restrict__ xl, const float* __restrict__ xr,
                                  const float* __restrict__ att, const float* __restrict__ bvec,
                                  const int* __restrict__ csr_off, const int* __restrict__ csr_src,
                                  float* __restrict__ out, int n) {
  const int lane = threadIdx.x & 31;
  const int node = blockIdx.x * (blockDim.x >> 5) + (threadIdx.x >> 5);
  if (node >= n) return;
  constexpr int DPL = (DOUT + 31) / 32;
  float xrc[DPL], attc[DPL];
#pragma unroll
  for (int dd = 0; dd < DPL; ++dd) {
    int idx = lane + dd * 32;
    bool ok = idx < DOUT;
    xrc[dd]  = ok ? xr[(long)node * DOUT + idx] : 0.0f;
    attc[dd] = ok ? att[idx] : 0.0f;
  }
  const int e0 = csr_off[node], e1 = csr_off[node + 1];
  // warm the first source row
  if (e0 < e1) __builtin_prefetch(xl + (long)csr_src[e0] * DOUT + lane, 0, 1);
  float emax = -1e30f;
  for (int e = e0; e <= e1; ++e) {  // e==e1 -> self loop
    const int s = (e < e1) ? csr_src[e] : node;
    if (e + 1 < e1)  // prefetch next edge's source row one iteration ahead
      __builtin_prefetch(xl + (long)csr_src[e + 1] * DOUT + lane, 0, 1);
    float p = 0.0f;
#pragma unroll
    for (int dd = 0; dd < DPL; ++dd) {
      int idx = lane + dd * 32;
      float v = (idx < DOUT) ? (xl[(long)s * DOUT + idx] + xrc[dd]) : 0.0f;
      v = (v > 0.0f) ? v : 0.2f * v;
      p += v * attc[dd];
    }
#pragma unroll
    for (int off = 16; off > 0; off >>= 1) p += __shfl_xor(p, off, 32);
    emax = fmaxf(emax, p);
  }
  float denom = 0.0f;
  float accv[DPL];
#pragma unroll
  for (int dd = 0; dd < DPL; ++dd) accv[dd] = 0.0f;
  for (int e = e0; e <= e1; ++e) {
    const int s = (e < e1) ? csr_src[e] : node;
    if (e + 1 < e1)
      __builtin_prefetch(xl + (long)csr_src[e + 1] * DOUT + lane, 0, 1);
    float xls[DPL];
    float p = 0.0f;
#pragma unroll
    for (int dd = 0; dd < DPL; ++dd) {
      int idx = lane + dd * 32;
      xls[dd] = (idx < DOUT) ? xl[(long)s * DOUT + idx] : 0.0f;
      float v = xls[dd] + xrc[dd];
      v = (v > 0.0f) ? v : 0.2f * v;
      p += v * attc[dd];
    }
#pragma unroll
    for (int off = 16; off > 0; off >>= 1) p += __shfl_xor(p, off, 32);
    float w = __expf(p - emax);
    denom += w;
#pragma unroll
    for (int dd = 0; dd < DPL; ++dd) accv[dd] += w * xls[dd];
  }
  const float inv = 1.0f / denom;
#pragma unroll
  for (int dd = 0; dd < DPL; ++dd) {
    int idx = lane + dd * 32;
    if (idx < DOUT) out[(long)node * DOUT + idx] = accv[dd] * inv + bvec[idx];
  }
}

// =====================================================================
//  Dense GATv2 attention matrix: alpha[b,i,:] = softmax_j( leaky(xr_i+xl_j).att )
//  One wave per (b,i); ncl <= 64.
// =====================================================================
template <int DOUT>
__global__ void gat_dense_alpha_kernel(const float* __restrict__ xl, const float* __restrict__ xr,
                                       const float* __restrict__ att, float* __restrict__ alpha,
                                       int nb, int ncl) {
  const int lane = threadIdx.x & 31;
  const int wid = blockIdx.x * (blockDim.x >> 5) + (threadIdx.x >> 5);
  if (wid >= nb * ncl) return;
  const int b = wid / ncl, i = wid - b * ncl;
  constexpr int DPL = (DOUT + 31) / 32;
  float xrc[DPL], attc[DPL];
  const long rowi = (long)(b * ncl + i) * DOUT;
#pragma unroll
  for (int dd = 0; dd < DPL; ++dd) {
    int idx = lane + dd * 32;
    bool ok = idx < DOUT;
    xrc[dd]  = ok ? xr[rowi + idx] : 0.0f;
    attc[dd] = ok ? att[idx] : 0.0f;
  }
  float ev0 = -1e30f, ev1 = -1e30f;
  for (int j = 0; j < ncl; ++j) {
    const long rowj = (long)(b * ncl + j) * DOUT;
    float p = 0.0f;
#pragma unroll
    for (int dd = 0; dd < DPL; ++dd) {
      int idx = lane + dd * 32;
      float v = (idx < DOUT) ? (xl[rowj + idx] + xrc[dd]) : 0.0f;
      v = (v > 0.0f) ? v : 0.2f * v;
      p += v * attc[dd];
    }
#pragma unroll
    for (int off = 16; off > 0; off >>= 1) p += __shfl_xor(p, off, 32);
    if ((j & 31) == lane) { if (j < 32) ev0 = p; else ev1 = p; }
  }
  float m = fmaxf(ev0, ev1);
#pragma unroll
  for (int off = 16; off > 0; off >>= 1) m = fmaxf(m, __shfl_xor(m, off, 32));
  float w0 = (lane < ncl)      ? __expf(ev0 - m) : 0.0f;
  float w1 = (lane + 32 < ncl) ? __expf(ev1 - m) : 0.0f;
  float s = w0 + w1;
#pragma unroll
  for (int off = 16; off > 0; off >>= 1) s += __shfl_xor(s, off, 32);
  const float inv = 1.0f / s;
  const long orow = (long)(b * ncl + i) * ncl;
  if (lane < ncl)      alpha[orow + lane]      = w0 * inv;
  if (lane + 32 < ncl) alpha[orow + lane + 32] = w1 * inv;
}

// =====================================================================
//  Host orchestration
// =====================================================================
static inline dim3 gblk(long total, int bs) { return dim3((unsigned)((total + bs - 1) / bs)); }

extern "C" void kernel_launch(void* const* d_in, const int* in_sizes, int n_in,
                              void* d_out, int out_size, void* d_ws, size_t ws_size,
                              hipStream_t stream) {
  (void)n_in; (void)out_size; (void)ws_size;
  // ---- inputs (setup_inputs dict order, params leaves in insertion order) ----
  const float* x        = (const float*)d_in[0];
  const int*   eidx     = (const int*)d_in[1];
  const int*   batch    = (const int*)d_in[2];
  const float* g1p_Wl  = (const float*)d_in[3];
  const float* g1p_bl  = (const float*)d_in[4];
  const float* g1p_Wr  = (const float*)d_in[5];
  const float* g1p_br  = (const float*)d_in[6];
  const float* g1p_att = (const float*)d_in[7];
  const float* g1p_b   = (const float*)d_in[8];
  const float* g1e_Wl  = (const float*)d_in[9];
  const float* g1e_bl  = (const float*)d_in[10];
  const float* g1e_Wr  = (const float*)d_in[11];
  const float* g1e_br  = (const float*)d_in[12];
  const float* g1e_att = (const float*)d_in[13];
  const float* g1e_b   = (const float*)d_in[14];
  const float* g2p_Wl  = (const float*)d_in[15];
  const float* g2p_bl  = (const float*)d_in[16];
  const float* g2p_Wr  = (const float*)d_in[17];
  const float* g2p_br  = (const float*)d_in[18];
  const float* g2p_att = (const float*)d_in[19];
  const float* g2p_b   = (const float*)d_in[20];
  const float* g2e_Wl  = (const float*)d_in[21];
  const float* g2e_bl  = (const float*)d_in[22];
  const float* g2e_Wr  = (const float*)d_in[23];
  const float* g2e_br  = (const float*)d_in[24];
  const float* g2e_att = (const float*)d_in[25];
  const float* g2e_b   = (const float*)d_in[26];
  const float* fp_W    = (const float*)d_in[27];
  const float* fp_b    = (const float*)d_in[28];
  float* outp = (float*)d_out;

  const int n  = in_sizes[0] / DIN;   // 32768 nodes
  const int ne = in_sizes[1] / 2;     // 524288 edges
  const int* srcE = eidx;
  const int* dstE = eidx + ne;

  // ---- workspace arena (byte offsets, phase-aliased) ----
  char* ws = (char*)d_ws;
  const size_t o_xl1e = 0;                 // 33,554,432   (phase2: x1e dense)
  const size_t o_xr1e = 33554432u;         // 33,554,432   (phase2+: x1 / alpha buffers)
  const size_t o_xl1p = 67108864u;         // 8,388,608    (phase2: s_d)
  const size_t o_xr1p = 75497472u;         // 8,388,608
  const size_t o_x1eo = 83886080u;         // 33,554,432   (phase3 pack)
  const size_t o_s1o  = 117440512u;        // 8,388,608
  const size_t o_int  = 125829120u;

  float* xl1e  = (float*)(ws + o_xl1e);
  float* xr1e  = (float*)(ws + o_xr1e);
  float* xl1p  = (float*)(ws + o_xl1p);
  float* xr1p  = (float*)(ws + o_xr1p);
  float* x1eo  = (float*)(ws + o_x1eo);
  float* s1o   = (float*)(ws + o_s1o);
  // phase-2 aliases
  float* x1e_d = (float*)(ws + o_xl1e);                 // [NB, NMAX, HID]
  float* s_d   = (float*)(ws + o_xl1p);                 // [NB, NMAX, C1]
  float* x1    = (float*)(ws + o_xr1e);                 // [NB, C1, HID]   4,194,304 B
  float* al2e  = (float*)(ws + o_xr1e + 4194304u);      // [NB, 64, 64]    1,048,576 B
  float* al2p  = (float*)(ws + o_xr1e + 5242880u);      // [NB, 64, 64]
  // phase-3 pack (over x1eo region)
  float* xl2e  = (float*)(ws + o_x1eo);                 // [4096, 256]
  float* xr2e  = (float*)(ws + o_x1eo + 4194304u);
  float* x2eo  = (float*)(ws + o_x1eo + 8388608u);      // [NB, 64, 256]
  float* xl2p  = (float*)(ws + o_x1eo + 12582912u);     // [4096, 16]
  float* xr2p  = (float*)(ws + o_x1eo + 12845056u);
  float* s2o   = (float*)(ws + o_x1eo + 13107200u);
  float* s2s   = (float*)(ws + o_x1eo + 13369344u);
  float* x2    = (float*)(ws + o_x1eo + 13631488u);     // [NB, 16, 256]
  float* emb   = (float*)(ws + o_x1eo + 14680064u);     // [NB, 768]
  // integer arrays
  int* counts  = (int*)(ws + o_int);
  int* starts  = (int*)(ws + o_int + 1024);
  int* localv  = (int*)(ws + o_int + 2048);
  int* deg     = (int*)(ws + o_int + 133120u);
  int* offs    = (int*)(ws + o_int + 264192u);
  int* cursor  = (int*)(ws + o_int + 395328u);
  int* csr_src = (int*)(ws + o_int + 526400u);

  // ---- phase 0: graph structure ----
  fill_i32_kernel<<<gblk(NB, 256), 256, 0, stream>>>(counts, 0, NB);
  count_batch_kernel<<<gblk(n, 256), 256, 0, stream>>>(batch, counts, n);
  scan64_kernel<<<1, 32, 0, stream>>>(counts, starts, NB);
  local_kernel<<<gblk(n, 256), 256, 0, stream>>>(batch, starts, localv, n);
  fill_i32_kernel<<<gblk(n, 256), 256, 0, stream>>>(deg, 0, n);
  count_deg_kernel<<<gblk(ne, 256), 256, 0, stream>>>(dstE, deg, ne);
  scan_offsets_kernel<<<1, 1024, 0, stream>>>(deg, offs, n);
  fill_i32_kernel<<<gblk(n, 256), 256, 0, stream>>>(cursor, 0, n);
  csr_fill_kernel<<<gblk(ne, 256), 256, 0, stream>>>(srcE, dstE, offs, cursor, csr_src, ne);

  // ---- phase 1: node linear transforms (WMMA) ----
  run_gemm(stream, false, true, x, g1e_Wl, g1e_bl, xl1e, n, HIDD, DIN, DIN, DIN, HIDD, 0, 0, 0, 1);
  run_gemm(stream, false, true, x, g1e_Wr, g1e_br, xr1e, n, HIDD, DIN, DIN, DIN, HIDD, 0, 0, 0, 1);
  run_gemm(stream, false, true, x, g1p_Wl, g1p_bl, xl1p, n, C1D,  DIN, DIN, DIN, C1D,  0, 0, 0, 1);
  run_gemm(stream, false, true, x, g1p_Wr, g1p_br, xr1p, n, C1D,  DIN, DIN, DIN, C1D,  0, 0, 0, 1);

  // ---- sparse GATv2 (embed + pool) ----
  gat_sparse_kernel<HIDD><<<gblk(n, 4), 128, 0, stream>>>(xl1e, xr1e, g1e_att, g1e_b,
                                                          offs, csr_src, x1eo, n);
  gat_sparse_kernel<C1D><<<gblk(n, 4), 128, 0, stream>>>(xl1p, xr1p, g1p_att, g1p_b,
                                                         offs, csr_src, s1o, n);

  // ---- densify (zero padding slots, then scatter valid rows) ----
  fill_f32_kernel<<<gblk((long)NB * NMAX * HIDD, 256), 256, 0, stream>>>(x1e_d, 0.0f, (long)NB * NMAX * HIDD);
  fill_f32_kernel<<<gblk((long)NB * NMAX * C1D, 256), 256, 0, stream>>>(s_d, 0.0f, (long)NB * NMAX * C1D);
  scatter_rows_kernel<<<gblk((long)n * HIDD, 256), 256, 0, stream>>>(x1eo, batch, localv, x1e_d, n, HIDD, NMAX);
  softmax_scatter_s_kernel<<<gblk(n, 4), 128, 0, stream>>>(s1o, batch, localv, s_d, n, NMAX);

  // ---- diffpool-1: x1[b] = s_d[b]^T @ x1e_d[b]   (adj path is dead code) ----
  run_gemm(stream, true, false, s_d, x1e_d, nullptr, x1,
           C1D, HIDD, NMAX, C1D, HIDD, HIDD,
           (long)NMAX * C1D, (long)NMAX * HIDD, (long)C1D * HIDD, NB);

  // ---- gnn2 transforms on x1 (flattened [NB*C1, HID]) ----
  const int M2 = NB * C1D;  // 4096
  run_gemm(stream, false, true, x1, g2e_Wl, g2e_bl, xl2e, M2, HIDD, HIDD, HIDD, HIDD, HIDD, 0, 0, 0, 1);
  run_gemm(stream, false, true, x1, g2e_Wr, g2e_br, xr2e, M2, HIDD, HIDD, HIDD, HIDD, HIDD, 0, 0, 0, 1);
  run_gemm(stream, false, true, x1, g2p_Wl, g2p_bl, xl2p, M2, C2D,  HIDD, HIDD, HIDD, C2D,  0, 0, 0, 1);
  run_gemm(stream, false, true, x1, g2p_Wr, g2p_br, xr2p, M2, C2D,  HIDD, HIDD, HIDD, C2D,  0, 0, 0, 1);

  // ---- dense GATv2 (complete graph over C1 clusters) ----
  gat_dense_alpha_kernel<HIDD><<<gblk(NB * C1D, 4), 128, 0, stream>>>(xl2e, xr2e, g2e_att, al2e, NB, C1D);
  run_gemm(stream, false, false, al2e, xl2e, g2e_b, x2eo,
           C1D, HIDD, C1D, C1D, HIDD, HIDD,
           (long)C1D * C1D, (long)C1D * HIDD, (long)C1D * HIDD, NB);
  gat_dense_alpha_kernel<C2D><<<gblk(NB * C1D, 4), 128, 0, stream>>>(xl2p, xr2p, g2p_att, al2p, NB, C1D);
  run_gemm(stream, false, false, al2p, xl2p, g2p_b, s2o,
           C1D, C2D, C1D, C1D, C2D, C2D,
           (long)C1D * C1D, (long)C1D * C2D, (long)C1D * C2D, NB);

  // ---- diffpool-2: softmax(s2) then x2[b] = s2[b]^T @ x2e[b] ----
  softmax_row_kernel<<<gblk(NB * C1D, 256), 256, 0, stream>>>(s2o, s2s, NB * C1D, C2D);
  run_gemm(stream, true, false, s2s, x2eo, nullptr, x2,
           C2D, HIDD, C1D, C2D, HIDD, HIDD,
           (long)C1D * C2D, (long)C1D * HIDD, (long)C2D * HIDD, NB);

  // ---- pooled embedding [pool0 | mean(x1) | mean(x2)] ----
  pool_mean_x_kernel<<<NB, DIN, 0, stream>>>(x, starts, counts, emb, DIN, 768, 0);
  mean_rows_kernel<<<NB, HIDD, 0, stream>>>(x1, emb, C1D, HIDD, 768, 256);
  mean_rows_kernel<<<NB, HIDD, 0, stream>>>(x2, emb, C2D, HIDD, 768, 512);

  // ---- final projection: out = emb @ W^T + b ----
  run_gemm(stream, false, true, emb, fp_W, fp_b, outp,
           NB, ODIM, 768, 768, 768, ODIM, 0, 0, 0, 1);
}